// HRM_5815385719046
// MI455X (gfx1250) — compile-verified
//
#include <hip/hip_runtime.h>
#include <math.h>

// ---------------- problem constants ----------------
#define DD 2048
#define HH 4096
#define KSTEP 4
#define NSTEPS 16
#define MOMC 0.9f
#define THRC 0.01f

// ---------------- workspace layout (float offsets) ----------------
#define O_H      0                    // h state           [HH]
#define O_L      (O_H + HH)           // l state           [DD]
#define O_RESULT (O_L + DD)           // result accum      [DD]
#define O_DIR    (O_RESULT + DD)      // directive         [DD]
#define O_RES    (O_DIR + DD)         // res               [DD]
#define O_GI     (O_RES + DD)         // gi scratch        [3*HH]
#define O_GH     (O_GI + 3*HH)        // gh scratch        [3*HH]
#define O_XCAT   (O_GH + 3*HH)        // concat(q,l)       [2*DD]
#define O_SCAL   (O_XCAT + 2*DD)      // scalars           [8]
#define O_FLAG   (O_SCAL + 8)         // int flags region  [8 ints]

#define S_CUM    0
#define S_WEIGHT 1
#define S_BEST   2
#define S_MOM    3
#define S_P      4

#define F_ACTIVE 0
#define F_RESET  1

typedef float v2f __attribute__((ext_vector_type(2)));
typedef float v4f __attribute__((ext_vector_type(4)));
typedef float v8f __attribute__((ext_vector_type(8)));

__device__ __forceinline__ float sigf(float x) { return 1.0f / (1.0f + expf(-x)); }

// ---------------------------------------------------------------------------
// init: zero h, l, result; reset scalar state and flags. 16 blocks x 256.
// ---------------------------------------------------------------------------
__global__ void init_state(float* __restrict__ ws) {
    int i = blockIdx.x * 256 + threadIdx.x;
    if (i < HH) ws[O_H + i] = 0.0f;
    if (i < DD) { ws[O_L + i] = 0.0f; ws[O_RESULT + i] = 0.0f; }
    if (i == 0) {
        ws[O_SCAL + S_CUM] = 0.0f; ws[O_SCAL + S_WEIGHT] = 0.0f;
        ws[O_SCAL + S_BEST] = 0.0f; ws[O_SCAL + S_MOM] = 0.0f;
        ws[O_SCAL + S_P] = 0.0f;
        int* fl = (int*)(ws + O_FLAG);
        fl[F_ACTIVE] = 1; fl[F_RESET] = 0;
    }
}

// ---------------------------------------------------------------------------
// WMMA fp32 matvec: y[r] = act( sum_k W[r,k]*x[k] + b[r] )
// Block = 8 waves, 16 rows; waves split K contiguously (scalar bounds via
// readfirstlane -> scalar loop control + real unrolling). x staged in LDS.
// Steady state per 8 K-elements:
//   1x global_load_b128 (W row chunk)  -> A operands {w.xy},{w.zw}
//   1x b128 LDS read    (x, broadcast) -> B operands {x.xy},{x.zw}
//   2x v_wmma_f32_16x16x4_f32 (accumulator chain)
// A and B use the same (VGPR, lane-half) -> K-slot map, so the pairing is
// correct under any hardware slot labeling. B constant across each half's 16
// lanes -> D columns identical; lanes 0/16 carry the 16 row sums.
// act: 0 = none, 1 = tanh.  gate: uniform early-exit when *gate == 0.
// ---------------------------------------------------------------------------
__global__ __launch_bounds__(256) void matvec_wmma(
    const float* __restrict__ W, const float* __restrict__ x,
    const float* __restrict__ b, float* __restrict__ y,
    int N, int act, const int* __restrict__ gate) {
    if (gate && gate[0] == 0) return;
    __shared__ float lds_x[HH];          // max N = 4096 floats = 16 KB
    __shared__ float partial[8 * 16];

    // stage x into LDS, coalesced b128 per thread
    for (int i = threadIdx.x * 4; i < N; i += 256 * 4)
        *(v4f*)(lds_x + i) = *(const v4f*)(x + i);
    __syncthreads();

    const int lane = threadIdx.x & 31;
    // wave id is wave-uniform: force it into an SGPR so the K-loop bounds are
    // scalar -> s_cmp/s_cbranch loop + unrolling instead of exec-masked loop.
    const int wave = __builtin_amdgcn_readfirstlane(threadIdx.x >> 5);
    const int half = lane >> 4;              // 0: cols k..k+3, 1: cols k+4..k+7
    const int r0   = blockIdx.x * 16;
    const int row  = r0 + (lane & 15);
    const int span = N >> 3;                 // N / 8 waves
    const int k0   = wave * span;
    const int k1   = k0 + span;

    const float* wrow = W + (size_t)row * (size_t)N;
    v8f c = {};

#pragma unroll 4
    for (int k = k0; k < k1; k += 8) {
        v4f w4 = *(const v4f*)(wrow + k + half * 4);    // A: 16B per lane
        v4f xb = *(const v4f*)(lds_x + k + half * 4);   // B: LDS broadcast
        v2f a0; a0.x = w4.x; a0.y = w4.y;
        v2f a1; a1.x = w4.z; a1.y = w4.w;
        v2f b0; b0.x = xb.x; b0.y = xb.y;
        v2f b1; b1.x = xb.z; b1.y = xb.w;
        c = __builtin_amdgcn_wmma_f32_16x16x4_f32(false, a0, false, b0,
                                                  (short)0, c, false, false);
        c = __builtin_amdgcn_wmma_f32_16x16x4_f32(false, a1, false, b1,
                                                  (short)0, c, false, false);
    }

    // C/D layout: VGPR v, lanes0-15 -> M=v, lanes16-31 -> M=8+v (cols equal)
    if ((lane & 15) == 0) {
        float* dst = &partial[wave * 16 + half * 8];
#pragma unroll
        for (int v = 0; v < 8; ++v) dst[v] = c[v];
    }
    __syncthreads();
    if (threadIdx.x < 16) {
        float s = 0.0f;
#pragma unroll
        for (int w = 0; w < 8; ++w) s += partial[w * 16 + threadIdx.x];
        s += b[r0 + threadIdx.x];
        if (act == 1) s = tanhf(s);
        y[r0 + threadIdx.x] = s;
    }
}

// ---------------------------------------------------------------------------
// xcat = concat(q, l)   (gated on active; only consumed by gated h_gru matvec)
// ---------------------------------------------------------------------------
__global__ void pack_xcat(const float* __restrict__ q, const float* __restrict__ l,
                          float* __restrict__ xcat, const int* __restrict__ gate) {
    if (gate[0] == 0) return;
    int i = blockIdx.x * 256 + threadIdx.x;
    if (i < DD) xcat[i] = q[i];
    else if (i < 2 * DD) xcat[i] = l[i - DD];
}

// ---------------------------------------------------------------------------
// GRU combine: v = (1-z)*n + z*v  from pre-activations gi, gh (biases included)
// ---------------------------------------------------------------------------
__global__ void gru_combine(const float* __restrict__ gi, const float* __restrict__ gh,
                            float* __restrict__ v, int n, const int* __restrict__ gate) {
    if (gate[0] == 0) return;
    int i = blockIdx.x * 256 + threadIdx.x;
    if (i >= n) return;
    float r  = sigf(gi[i]         + gh[i]);
    float z  = sigf(gi[n + i]     + gh[n + i]);
    float nn = tanhf(gi[2 * n + i] + r * gh[2 * n + i]);
    v[i] = (1.0f - z) * nn + z * v[i];
}

// ---------------------------------------------------------------------------
// conv/mom: raw = sigmoid(h_conv_w . h + b); mom = MOM*mom + (1-MOM)*raw (if active)
// ---------------------------------------------------------------------------
__global__ void conv_mom(const float* __restrict__ w, const float* __restrict__ b,
                         const float* __restrict__ h, float* __restrict__ scal,
                         const int* __restrict__ flags) {
    if (flags[F_ACTIVE] == 0) return;
    __shared__ float red[256];
    float s = 0.0f;
    for (int i = threadIdx.x; i < HH; i += 256) s += w[i] * h[i];
    red[threadIdx.x] = s; __syncthreads();
    for (int o = 128; o > 0; o >>= 1) {
        if (threadIdx.x < o) red[threadIdx.x] += red[threadIdx.x + o];
        __syncthreads();
    }
    if (threadIdx.x == 0) {
        float raw = sigf(red[0] + b[0]);
        scal[S_MOM] = MOMC * scal[S_MOM] + (1.0f - MOMC) * raw;
    }
}

// ---------------------------------------------------------------------------
// halt + reset + scalar state update (one block). Uses h,l BEFORE l_init reset.
// ---------------------------------------------------------------------------
__global__ void halt_reset(const float* __restrict__ hw, const float* __restrict__ hb,
                           const float* __restrict__ rw, const float* __restrict__ rb,
                           const float* __restrict__ h, const float* __restrict__ l,
                           float* __restrict__ scal, int* __restrict__ flags, int step) {
    __shared__ float r1[256], r2[256];
    float s1 = 0.0f, s2 = 0.0f;
    for (int i = threadIdx.x; i < HH; i += 256) { float v = h[i]; s1 += hw[i] * v; s2 += rw[i] * v; }
    for (int i = threadIdx.x; i < DD; i += 256) { float v = l[i]; s1 += hw[HH + i] * v; s2 += rw[HH + i] * v; }
    r1[threadIdx.x] = s1; r2[threadIdx.x] = s2; __syncthreads();
    for (int o = 128; o > 0; o >>= 1) {
        if (threadIdx.x < o) { r1[threadIdx.x] += r1[threadIdx.x + o]; r2[threadIdx.x] += r2[threadIdx.x + o]; }
        __syncthreads();
    }
    if (threadIdx.x == 0) {
        int   active = flags[F_ACTIVE];
        float conv   = scal[S_MOM];
        float halt   = sigf(r1[0] + hb[0]);
        float cum    = scal[S_CUM];
        float p      = active ? fminf(halt, 1.0f - cum) : 0.0f;
        scal[S_P] = p;
        cum += p;
        scal[S_CUM]    = cum;
        scal[S_WEIGHT] = scal[S_WEIGHT] + p;
        if (active && conv > scal[S_BEST] + THRC) scal[S_BEST] = conv;
        int break_now = (cum > 0.95f) ? 1 : 0;
        float rp = sigf(r2[0] + rw[HH + DD] * conv + rb[0]);
        flags[F_RESET]  = (active && !break_now && (rp > 0.7f) && (step > KSTEP)) ? 1 : 0;
        flags[F_ACTIVE] = (active && !break_now) ? 1 : 0;
    }
}

// ---------------------------------------------------------------------------
// result += p * res  (p == 0 when inactive -> no-op)
// ---------------------------------------------------------------------------
__global__ void axpy_result(const float* __restrict__ res, const float* __restrict__ scal,
                            float* __restrict__ result) {
    int i = blockIdx.x * 256 + threadIdx.x;
    if (i < DD) result[i] += scal[S_P] * res[i];
}

// out = result / max(weight, 1e-8)
__global__ void finalize(const float* __restrict__ result, const float* __restrict__ scal,
                         float* __restrict__ out) {
    int i = blockIdx.x * 256 + threadIdx.x;
    if (i < DD) out[i] = result[i] / fmaxf(scal[S_WEIGHT], 1e-8f);
}

// ---------------------------------------------------------------------------
extern "C" void kernel_launch(void* const* d_in, const int* in_sizes, int n_in,
                              void* d_out, int out_size, void* d_ws, size_t ws_size,
                              hipStream_t stream) {
    (void)in_sizes; (void)n_in; (void)out_size; (void)ws_size;
    const float* query     = (const float*)d_in[0];
    const float* h_gru_wih = (const float*)d_in[1];
    const float* h_gru_whh = (const float*)d_in[2];
    const float* h_gru_bih = (const float*)d_in[3];
    const float* h_gru_bhh = (const float*)d_in[4];
    const float* l_gru_wih = (const float*)d_in[5];
    const float* l_gru_whh = (const float*)d_in[6];
    const float* l_gru_bih = (const float*)d_in[7];
    const float* l_gru_bhh = (const float*)d_in[8];
    const float* h_dir_w   = (const float*)d_in[9];
    const float* h_dir_b   = (const float*)d_in[10];
    const float* h_conv_w  = (const float*)d_in[11];
    const float* h_conv_b  = (const float*)d_in[12];
    const float* l_out_w   = (const float*)d_in[13];
    const float* l_out_b   = (const float*)d_in[14];
    const float* halt_w    = (const float*)d_in[15];
    const float* halt_b    = (const float*)d_in[16];
    const float* reset_w   = (const float*)d_in[17];
    const float* reset_b   = (const float*)d_in[18];
    const float* l_init_w  = (const float*)d_in[19];
    const float* l_init_b  = (const float*)d_in[20];

    float* WS     = (float*)d_ws;
    float* h      = WS + O_H;
    float* l      = WS + O_L;
    float* result = WS + O_RESULT;
    float* dir    = WS + O_DIR;
    float* res    = WS + O_RES;
    float* gi     = WS + O_GI;
    float* gh     = WS + O_GH;
    float* xcat   = WS + O_XCAT;
    float* scal   = WS + O_SCAL;
    int*   flags  = (int*)(WS + O_FLAG);
    const int* gA = flags + F_ACTIVE;
    const int* gR = flags + F_RESET;

    init_state<<<16, 256, 0, stream>>>(WS);

    for (int t = 0; t < NSTEPS; ++t) {
        if (t % KSTEP == 0) {
            pack_xcat<<<(2 * DD) / 256, 256, 0, stream>>>(query, l, xcat, gA);
            matvec_wmma<<<(3 * HH) / 16, 256, 0, stream>>>(h_gru_wih, xcat, h_gru_bih, gi, 2 * DD, 0, gA);
            matvec_wmma<<<(3 * HH) / 16, 256, 0, stream>>>(h_gru_whh, h,    h_gru_bhh, gh, HH,     0, gA);
            gru_combine<<<HH / 256, 256, 0, stream>>>(gi, gh, h, HH, gA);
        }
        matvec_wmma<<<DD / 16, 256, 0, stream>>>(h_dir_w, h, h_dir_b, dir, HH, 1, gA);
        conv_mom<<<1, 256, 0, stream>>>(h_conv_w, h_conv_b, h, scal, flags);
        matvec_wmma<<<(3 * DD) / 16, 256, 0, stream>>>(l_gru_wih, dir, l_gru_bih, gi, DD, 0, gA);
        matvec_wmma<<<(3 * DD) / 16, 256, 0, stream>>>(l_gru_whh, l,   l_gru_bhh, gh, DD, 0, gA);
        gru_combine<<<DD / 256, 256, 0, stream>>>(gi, gh, l, DD, gA);
        matvec_wmma<<<DD / 16, 256, 0, stream>>>(l_out_w, l, l_out_b, res, DD, 1, gA);
        halt_reset<<<1, 256, 0, stream>>>(halt_w, halt_b, reset_w, reset_b, h, l, scal, flags, t);
        axpy_result<<<DD / 256, 256, 0, stream>>>(res, scal, result);
        matvec_wmma<<<DD / 16, 256, 0, stream>>>(l_init_w, h, l_init_b, l, HH, 1, gR);
    }

    finalize<<<DD / 256, 256, 0, stream>>>(result, scal, (float*)d_out);
}